// DBLoss_32074815766655
// MI455X (gfx1250) — compile-verified
//
#include <hip/hip_runtime.h>
#include <math.h>

// ---------------- problem constants (match reference) ----------------
#define NUM_CLASSES 80
#define BB 16
#define NA 3
#define HH 80
#define WW 80
#define DD 85          // 5 + NUM_CLASSES
#define NLAB 48
#define NCELL (BB*NA*HH*WW)      // 307200
#define MAIN_BLOCKS (NCELL/256)  // 1200
#define PI_F 3.14159265358979323846f

typedef float v2f __attribute__((ext_vector_type(2)));
typedef float v8f __attribute__((ext_vector_type(8)));

__device__ __forceinline__ float softplusf(float x) {
    // matches jax.nn.softplus numerics: max(x,0) + log1p(exp(-|x|))
    return fmaxf(x, 0.f) + log1pf(expf(-fabsf(x)));
}
__device__ __forceinline__ float sigmoidf_(float x) {
    return 1.f / (1.f + expf(-x));
}

// ---------------- kernel 1: clear the per-cell target mask ----------------
__global__ void init_mask_kernel(uint4* __restrict__ mask) {
    int c = blockIdx.x * blockDim.x + threadIdx.x;
    if (c < NCELL) mask[c] = make_uint4(0u, 0u, 0u, 0u);
}

// ---------------- kernel 2: sequential per-batch target assignment ----------------
// One thread per batch reproduces the lax.scan ordering exactly:
//  - class bits accumulate (multi-hot, never cleared)
//  - box target is last-label-wins
__global__ void assign_targets_kernel(const float* __restrict__ lxywh,
                                      const int*   __restrict__ lcls,
                                      uint4*  __restrict__ mask,
                                      float4* __restrict__ tbox) {
    int b = threadIdx.x;
    if (b >= BB) return;
    const float aw[3] = {10.f/8.f, 16.f/8.f, 33.f/8.f};
    const float ah[3] = {13.f/8.f, 30.f/8.f, 23.f/8.f};
    for (int n = 0; n < NLAB; ++n) {
        const float* L = lxywh + ((size_t)b * NLAB + n) * 4;
        float gx = L[0] * 640.f, gy = L[1] * 640.f;
        float gw = L[2] * 640.f, gh = L[3] * 640.f;
        int gi = (int)fminf(fmaxf(gx / 8.f, 0.f), (float)WW - 0.001f);
        int gj = (int)fminf(fmaxf(gy / 8.f, 0.f), (float)HH - 0.001f);
        float gtw = gw / 8.f, gth = gh / 8.f;
        int best = 0; float bestr = -1.f;
        for (int a = 0; a < 3; ++a) {
            float inter = fminf(gtw, aw[a]) * fminf(gth, ah[a]);
            float uni   = gtw * gth + aw[a] * ah[a] - inter + 1e-9f;
            float r = inter / uni;
            if (r > bestr) { bestr = r; best = a; }   // strict >: first max wins
        }
        int c = lcls[b * NLAB + n];
        int word = c >> 5; unsigned bit = 1u << (c & 31);
        for (int di = -1; di <= 1; ++di) {
            for (int dj = -1; dj <= 1; ++dj) {
                int ii = min(max(gi + di, 0), WW - 1);
                int jj = min(max(gj + dj, 0), HH - 1);
                int cell = ((b * NA + best) * HH + jj) * WW + ii;
                uint4 m = mask[cell];
                if (word == 0) m.x |= bit; else if (word == 1) m.y |= bit; else m.z |= bit;
                m.w = 1u;                    // tobj = 1
                mask[cell] = m;
                tbox[cell] = make_float4(gx, gy, gw, gh);   // last label wins
            }
        }
    }
}

// ---------------- kernel 3: main streaming pass (one thread per cell) ----------------
__global__ void main_loss_kernel(const float*  __restrict__ p,
                                 const uint4*  __restrict__ mask,
                                 const float4* __restrict__ tbox,
                                 float4* __restrict__ partials) {
    const int cell = blockIdx.x * blockDim.x + threadIdx.x;   // exactly NCELL threads
    float sobj = 0.f, sbox = 0.f, scls = 0.f, npos = 0.f;
    {
        uint4 m = mask[cell];
        const float* pc = p + (size_t)cell * DD;
        float obj = pc[4];
        sobj = softplusf(obj);                 // bce(obj, t) = softplus(o) - t*o
        if (m.w) {
            sobj -= obj;
            npos  = 1.f;
            const float* cp = pc + 5;
            __builtin_prefetch(cp, 0, 3);      // global_prefetch for the 80-class block
            // decode cell -> (b,a,j,i)
            int i = cell % WW;
            int j = (cell / WW) % HH;
            int a = (cell / (WW * HH)) % NA;
            const float ancw[3] = {10.f, 16.f, 33.f};
            const float anch[3] = {13.f, 30.f, 23.f};
            float px = ((float)i + sigmoidf_(pc[0])) * 8.f;
            float py = ((float)j + sigmoidf_(pc[1])) * 8.f;
            float pw = expf(pc[2]) * ancw[a];  // (exp * anchors/stride) * stride
            float ph = expf(pc[3]) * anch[a];
            float4 t = tbox[cell];
            float gx = t.x, gy = t.y, gw = t.z, gh = t.w;
            // ---- CIoU (forward; stop_gradient irrelevant) ----
            float px1 = px - pw * 0.5f, py1 = py - ph * 0.5f;
            float px2 = px + pw * 0.5f, py2 = py + ph * 0.5f;
            float gx1 = gx - gw * 0.5f, gy1 = gy - gh * 0.5f;
            float gx2 = gx + gw * 0.5f, gy2 = gy + gh * 0.5f;
            float iw = fmaxf(fminf(px2, gx2) - fmaxf(px1, gx1), 0.f);
            float ih = fmaxf(fminf(py2, gy2) - fmaxf(py1, gy1), 0.f);
            float inter  = iw * ih;
            float area_p = fmaxf(px2 - px1, 0.f) * fmaxf(py2 - py1, 0.f);
            float area_g = fmaxf(gx2 - gx1, 0.f) * fmaxf(gy2 - gy1, 0.f);
            float uni = area_p + area_g - inter + 1e-7f;
            float iou = inter / uni;
            float cw = fmaxf(fmaxf(px2, gx2) - fminf(px1, gx1), 0.f);
            float ch = fmaxf(fmaxf(py2, gy2) - fminf(py1, gy1), 0.f);
            float c2 = cw * cw + ch * ch + 1e-7f;
            float rho2 = (px - gx) * (px - gx) + (py - gy) * (py - gy);
            float dat = atanf(gw / (gh + 1e-7f)) - atanf(pw / (ph + 1e-7f));
            float v = (4.f / (PI_F * PI_F)) * dat * dat;
            float alpha = v / (1.f - iou + v + 1e-7f);
            sbox = 1.f - (iou - rho2 / c2 - alpha * v);
            // ---- class BCE over multi-hot target ----
            float sc = 0.f;
            unsigned mw0 = m.x, mw1 = m.y, mw2 = m.z;
            for (int k = 0; k < NUM_CLASSES; ++k) {
                float x = cp[k];
                sc += softplusf(x);
                unsigned bits = (k < 32) ? mw0 : (k < 64) ? mw1 : mw2;
                if ((bits >> (k & 31)) & 1u) sc -= x;
            }
            scls = sc;
        }
    }
    // deterministic block reduction (LDS tree)
    __shared__ float4 red[256];
    red[threadIdx.x] = make_float4(sobj, sbox, scls, npos);
    __syncthreads();
    for (int s = 128; s > 0; s >>= 1) {
        if (threadIdx.x < s) {
            float4 a0 = red[threadIdx.x], b0 = red[threadIdx.x + s];
            red[threadIdx.x] = make_float4(a0.x + b0.x, a0.y + b0.y,
                                           a0.z + b0.z, a0.w + b0.w);
        }
        __syncthreads();
    }
    if (threadIdx.x == 0) partials[blockIdx.x] = red[0];
}

// ---------------- WMMA 32-lane sum: D = A x ones + 0 ----------------
// A (16x4, f32): lane L holds row M=L%16; VGPR0 is K=0 (lanes 0-15) / K=2 (lanes 16-31).
// Putting the lane value in a[0] and 0 in a[1] gives rowsum_m = v_m + v_{m+16}.
// D layout: VGPR r holds rows r (lanes 0-15) and r+8 (lanes 16-31), so
// sum of the 8 D regs per lane + shfl_xor(16) = exact 32-lane total.
__device__ __forceinline__ float wave_sum_wmma(float v) {
    v2f a; a[0] = v;  a[1] = 0.f;
    v2f b; b[0] = 1.f; b[1] = 1.f;       // B matrix = all ones
    v8f c = {};
    c = __builtin_amdgcn_wmma_f32_16x16x4_f32(
            /*neg_a=*/false, a, /*neg_b=*/false, b,
            /*c_mod=*/(short)0, c, /*reuse_a=*/false, /*reuse_b=*/false);
    float t = c[0] + c[1] + c[2] + c[3] + c[4] + c[5] + c[6] + c[7];
    t += __shfl_xor(t, 16, 32);
    return t;
}

// ---------------- kernel 4: final fixed-order reduce + loss assembly ----------------
__global__ void final_reduce_kernel(const float4* __restrict__ partials,
                                    int P, float* __restrict__ out) {
    int lane = threadIdx.x;     // single wave of 32, all lanes active at WMMA
    float s0 = 0.f, s1 = 0.f, s2 = 0.f, s3 = 0.f;
    for (int i = lane; i < P; i += 32) {
        float4 v = partials[i];
        s0 += v.x; s1 += v.y; s2 += v.z; s3 += v.w;
    }
    float t_obj  = wave_sum_wmma(s0);
    float t_box  = wave_sum_wmma(s1);
    float t_cls  = wave_sum_wmma(s2);
    float t_npos = wave_sum_wmma(s3);
    if (lane == 0) {
        float n_pos = fmaxf(t_npos, 1.f);
        float l_box = t_box / n_pos;
        float l_obj = t_obj / (float)NCELL;
        float l_cls = t_cls / (n_pos * (float)NUM_CLASSES);
        out[0] = 7.5f * l_box + 1.0f * l_obj + 0.5f * l_cls;
    }
}

// ---------------- launcher ----------------
extern "C" void kernel_launch(void* const* d_in, const int* in_sizes, int n_in,
                              void* d_out, int out_size, void* d_ws, size_t ws_size,
                              hipStream_t stream) {
    (void)in_sizes; (void)n_in; (void)out_size; (void)ws_size;
    const float* p_raw = (const float*)d_in[0];
    const float* lxywh = (const float*)d_in[1];
    const int*   lcls  = (const int*)  d_in[2];

    char* ws = (char*)d_ws;
    uint4*  mask     = (uint4*) ws;                               // NCELL*16 B
    float4* tboxp    = (float4*)(ws + (size_t)NCELL * 16);        // NCELL*16 B
    float4* partials = (float4*)(ws + (size_t)NCELL * 32);        // MAIN_BLOCKS*16 B

    init_mask_kernel   <<<MAIN_BLOCKS, 256, 0, stream>>>(mask);
    assign_targets_kernel<<<1, 16, 0, stream>>>(lxywh, lcls, mask, tboxp);
    main_loss_kernel   <<<MAIN_BLOCKS, 256, 0, stream>>>(p_raw, mask, tboxp, partials);
    final_reduce_kernel<<<1, 32, 0, stream>>>(partials, MAIN_BLOCKS, (float*)d_out);
}